// EMDLoss_3444563771755
// MI455X (gfx1250) — compile-verified
//
#include <hip/hip_runtime.h>

// Problem constants (match reference setup_inputs / hyperparams)
#define B_      8
#define N_      2048
#define EPS_    0.05f
#define ITERS_  100
#define MU_     (1.0f / (float)N_)

#define TILES_PER_DIM   (N_ / 16)                        // 128
#define TILES_PER_BATCH (TILES_PER_DIM * TILES_PER_DIM)  // 16384
#define NTILES          (B_ * TILES_PER_BATCH)           // 131072
#define WAVES_PER_BLOCK 4

typedef __attribute__((ext_vector_type(2))) float     v2f;
typedef __attribute__((ext_vector_type(8))) float     v8f;
typedef __attribute__((ext_vector_type(8))) _Float16  v8h;

// -----------------------------------------------------------------------------
// Compute a 16x16 tile of the Euclidean distance matrix via V_WMMA_F32_16X16X4_F32.
// A = template rows [16 x 4] (xyz padded with 0), B = source cols [4 x 16].
// d2 = |t|^2 + |s|^2 - 2 t.s ; dist = sqrt(max(d2, 1e-12)).
// C/D layout: VGPR r, lanes 0-15 -> (M=r, N=lane), lanes 16-31 -> (M=r+8, N=lane-16).
// -----------------------------------------------------------------------------
__device__ inline v8f dist_tile_wmma(const float* __restrict__ Tb,
                                     const float* __restrict__ Sb,
                                     const float* __restrict__ t2b,
                                     const float* __restrict__ s2b,
                                     int i0, int j0, int lane) {
    const int half = lane >> 4;   // 0: K={0,1}   1: K={2,3}
    const int l16  = lane & 15;

    // A fragment: row i0+l16, K components (2*half, 2*half+1); K=3 is zero pad.
    const int arow = i0 + l16;
    v2f a;
    a.x = Tb[arow * 3 + 2 * half];
    a.y = (half == 0) ? Tb[arow * 3 + 1] : 0.0f;

    // B fragment (mirrored layout, M<->N): column j0+l16, same K striping.
    const int bcol = j0 + l16;
    v2f b;
    b.x = Sb[bcol * 3 + 2 * half];
    b.y = (half == 0) ? Sb[bcol * 3 + 1] : 0.0f;

    v8f c = {0.f, 0.f, 0.f, 0.f, 0.f, 0.f, 0.f, 0.f};
    // 8 args: (neg_a, A, neg_b, B, c_mod, C, reuse_a, reuse_b)
    c = __builtin_amdgcn_wmma_f32_16x16x4_f32(false, a, false, b, (short)0, c,
                                              false, false);

    const float s2j = s2b[j0 + l16];
    v8f dist;
#pragma unroll
    for (int r = 0; r < 8; ++r) {
        const int i = i0 + r + 8 * half;
        const float d2 = t2b[i] + s2j - 2.0f * c[r];
        dist[r] = __builtin_amdgcn_sqrtf(fmaxf(d2, 1e-12f));
    }
    return dist;
}

// -----------------------------------------------------------------------------
// Init: squared norms of template/source points, b-scaling = 1.
// -----------------------------------------------------------------------------
__global__ void emd_init(const float* __restrict__ T, const float* __restrict__ S,
                         float* __restrict__ t2, float* __restrict__ s2,
                         float* __restrict__ bvec) {
    const int idx = blockIdx.x * blockDim.x + threadIdx.x;   // 0 .. B*N-1
    if (idx < B_ * N_) {
        float x = T[idx * 3 + 0], y = T[idx * 3 + 1], z = T[idx * 3 + 2];
        t2[idx] = x * x + y * y + z * z;
        x = S[idx * 3 + 0]; y = S[idx * 3 + 1]; z = S[idx * 3 + 2];
        s2[idx] = x * x + y * y + z * z;
        bvec[idx] = 1.0f;
    }
}

// -----------------------------------------------------------------------------
// Build K = exp(-dist/eps) (f16 storage: halves traffic; K+KT = 134 MB fits L2)
// and its explicit transpose (coalesced column pass). One wave per 16x16 tile;
// distances via WMMA.
// -----------------------------------------------------------------------------
__global__ void emd_build_kernelmat(const float* __restrict__ T,
                                    const float* __restrict__ S,
                                    const float* __restrict__ t2,
                                    const float* __restrict__ s2,
                                    _Float16* __restrict__ K,
                                    _Float16* __restrict__ KT) {
    const int wave = threadIdx.x >> 5;
    const int lane = threadIdx.x & 31;
    const int tile = blockIdx.x * WAVES_PER_BLOCK + wave;

    const int b   = tile / TILES_PER_BATCH;
    const int rem = tile % TILES_PER_BATCH;
    const int i0  = (rem / TILES_PER_DIM) * 16;
    const int j0  = (rem % TILES_PER_DIM) * 16;

    const float* Tb  = T  + (size_t)b * N_ * 3;
    const float* Sb  = S  + (size_t)b * N_ * 3;
    const float* t2b = t2 + (size_t)b * N_;
    const float* s2b = s2 + (size_t)b * N_;

    const v8f dist = dist_tile_wmma(Tb, Sb, t2b, s2b, i0, j0, lane);

    const int half = lane >> 4;
    const int j    = j0 + (lane & 15);
    const size_t base = (size_t)b * N_ * N_;
    const float inv_eps = 1.0f / EPS_;
#pragma unroll
    for (int r = 0; r < 8; ++r) {
        const int i = i0 + r + 8 * half;
        const _Float16 k = (_Float16)__expf(-dist[r] * inv_eps);
        K [base + (size_t)i * N_ + j] = k;
        KT[base + (size_t)j * N_ + i] = k;
    }
}

// -----------------------------------------------------------------------------
// One Sinkhorn half-step: scaleOut[row] = mu / (Kmat[row,:] . scaleIn).
// One 256-thread block per row; each thread loads 8 f16 (one b128) + 8 f32.
// Fixed-order LDS tree reduction (deterministic). Used for both directions
// (K with b -> a, KT with a -> b).
// -----------------------------------------------------------------------------
__global__ void emd_sinkhorn_step(const _Float16* __restrict__ Kmat,
                                  const float* __restrict__ scaleIn,
                                  float* __restrict__ scaleOut) {
    const int row = blockIdx.x;          // 0 .. B*N-1
    const int b   = row / N_;
    const int i   = row % N_;
    const _Float16* __restrict__ krow = Kmat + (size_t)b * N_ * N_ + (size_t)i * N_;
    const float* __restrict__ sv      = scaleIn + (size_t)b * N_;

    const int m0 = threadIdx.x * 8;      // 256 threads x 8 = 2048 = N_
    const v8h kv = *(const v8h*)(krow + m0);
    float acc = 0.0f;
#pragma unroll
    for (int r = 0; r < 8; ++r)
        acc += (float)kv[r] * sv[m0 + r];

    __shared__ float red[256];
    red[threadIdx.x] = acc;
    __syncthreads();
#pragma unroll
    for (int s = 128; s > 0; s >>= 1) {
        if (threadIdx.x < s) red[threadIdx.x] += red[threadIdx.x + s];
        __syncthreads();
    }
    if (threadIdx.x == 0)
        scaleOut[row] = MU_ / fmaxf(red[0], 1e-30f);
}

// -----------------------------------------------------------------------------
// Loss partials: per tile, sum a_i * K_ij * b_j * dist_ij (dist recomputed via
// WMMA — avoids NaN from log of underflowed K entries). One wave per tile.
// -----------------------------------------------------------------------------
__global__ void emd_loss_tiles(const float* __restrict__ T,
                               const float* __restrict__ S,
                               const float* __restrict__ t2,
                               const float* __restrict__ s2,
                               const _Float16* __restrict__ K,
                               const float* __restrict__ avec,
                               const float* __restrict__ bvec,
                               float* __restrict__ partial) {
    const int wave = threadIdx.x >> 5;
    const int lane = threadIdx.x & 31;
    const int tile = blockIdx.x * WAVES_PER_BLOCK + wave;

    const int b   = tile / TILES_PER_BATCH;
    const int rem = tile % TILES_PER_BATCH;
    const int i0  = (rem / TILES_PER_DIM) * 16;
    const int j0  = (rem % TILES_PER_DIM) * 16;

    const float* Tb  = T  + (size_t)b * N_ * 3;
    const float* Sb  = S  + (size_t)b * N_ * 3;
    const float* t2b = t2 + (size_t)b * N_;
    const float* s2b = s2 + (size_t)b * N_;

    const v8f dist = dist_tile_wmma(Tb, Sb, t2b, s2b, i0, j0, lane);

    const int half = lane >> 4;
    const int j    = j0 + (lane & 15);
    const size_t base = (size_t)b * N_ * N_;
    const float bj = bvec[(size_t)b * N_ + j];

    float acc = 0.0f;
#pragma unroll
    for (int r = 0; r < 8; ++r) {
        const int i = i0 + r + 8 * half;
        const float ai = avec[(size_t)b * N_ + i];
        acc += ai * (float)K[base + (size_t)i * N_ + j] * bj * dist[r];
    }
    // wave32 butterfly reduction (deterministic)
#pragma unroll
    for (int m = 16; m >= 1; m >>= 1)
        acc += __shfl_xor(acc, m, 32);
    if (lane == 0)
        partial[tile] = acc;
}

// -----------------------------------------------------------------------------
// Final reduction of tile partials -> scalar mean over batches.
// -----------------------------------------------------------------------------
__global__ void emd_reduce(const float* __restrict__ partial, float* __restrict__ out) {
    __shared__ float red[256];
    float acc = 0.0f;
    for (int idx = threadIdx.x; idx < NTILES; idx += 256)
        acc += partial[idx];
    red[threadIdx.x] = acc;
    __syncthreads();
#pragma unroll
    for (int s = 128; s > 0; s >>= 1) {
        if (threadIdx.x < s) red[threadIdx.x] += red[threadIdx.x + s];
        __syncthreads();
    }
    if (threadIdx.x == 0)
        out[0] = red[0] * (1.0f / (float)B_);
}

// -----------------------------------------------------------------------------
// Host launcher
// -----------------------------------------------------------------------------
extern "C" void kernel_launch(void* const* d_in, const int* in_sizes, int n_in,
                              void* d_out, int out_size, void* d_ws, size_t ws_size,
                              hipStream_t stream) {
    (void)in_sizes; (void)n_in; (void)out_size; (void)ws_size;

    const float* T = (const float*)d_in[0];   // [B, N, 3] f32
    const float* S = (const float*)d_in[1];   // [B, N, 3] f32
    float* out = (float*)d_out;               // scalar f32

    // Workspace layout (f16 kernel matrices: 67 MB each)
    char* ws = (char*)d_ws;
    const size_t szK = (size_t)B_ * N_ * N_ * sizeof(_Float16);
    _Float16* K  = (_Float16*)ws;
    _Float16* KT = (_Float16*)(ws + szK);
    float* t2  = (float*)(ws + 2 * szK);
    float* s2  = t2 + (size_t)B_ * N_;
    float* av  = s2 + (size_t)B_ * N_;
    float* bv  = av + (size_t)B_ * N_;
    float* prt = bv + (size_t)B_ * N_;        // NTILES floats

    // 1) norms + b = 1
    emd_init<<<(B_ * N_ + 255) / 256, 256, 0, stream>>>(T, S, t2, s2, bv);

    // 2) K = exp(-dist/eps) and KT (WMMA distance tiles)
    emd_build_kernelmat<<<NTILES / WAVES_PER_BLOCK, 32 * WAVES_PER_BLOCK, 0, stream>>>(
        T, S, t2, s2, K, KT);

    // 3) 100 Sinkhorn iterations (linear domain): a = mu/(K b); b = mu/(KT a)
    for (int it = 0; it < ITERS_; ++it) {
        emd_sinkhorn_step<<<B_ * N_, 256, 0, stream>>>(K,  bv, av);
        emd_sinkhorn_step<<<B_ * N_, 256, 0, stream>>>(KT, av, bv);
    }

    // 4) loss = mean_b sum_ij a_i K_ij b_j dist_ij
    emd_loss_tiles<<<NTILES / WAVES_PER_BLOCK, 32 * WAVES_PER_BLOCK, 0, stream>>>(
        T, S, t2, s2, K, av, bv, prt);
    emd_reduce<<<1, 256, 0, stream>>>(prt, out);
}